// SageModel_74526272520732
// MI455X (gfx1250) — compile-verified
//
#include <hip/hip_runtime.h>
#include <hip/hip_bf16.h>

// ---------------------------------------------------------------------------
// GraphSAGE 2-layer forward for MI455X (gfx1250, wave32).
//
//  Aggregation: CSR build (int histogram + scan + bin) then one wave per dst
//  node gather-reduces its neighbor rows in registers (coalesced b128 loads,
//  NO float atomics) and writes the mean.
//
//  GEMMs: fused concat([h_dst, h_N]) @ W via V_WMMA_F32_16X16X4_F32.
//  W is prepacked once per launch into B-fragment order; the K loop is split
//  at the concat boundary so every load in the hot loop is base+immediate:
//  1x b64 (A) + 2x b128 (4 B fragments) per 4 WMMAs.
// ---------------------------------------------------------------------------

typedef __attribute__((ext_vector_type(2))) float v2f;
typedef __attribute__((ext_vector_type(8))) float v8f;

#define N_DST1   50000
#define N_DST2   10000
#define IN_F     256      // per-half feature width (concat -> K = 512)
#define KDIM     512
#define KSTEPS   (KDIM / 4)     // 128 WMMA K-steps
#define KSPLIT   (IN_F / 4)     // 64: K-step where A switches from A0 to A1
#define HID_F    256      // layer-1 output width
#define N_CLS    64       // layer-2 output width

// ---------------------------------------------------------------------------
__global__ void zero_i32_kernel(int* __restrict__ p, int n) {
  int i = blockIdx.x * blockDim.x + threadIdx.x;
  if (i < n) p[i] = 0;
}

// histogram of destination indices
__global__ void hist_kernel(const int* __restrict__ dst,
                            int* __restrict__ hist, int nE) {
  int i = blockIdx.x * blockDim.x + threadIdx.x;
  if (i < nE) atomicAdd(&hist[dst[i]], 1);
}

// exclusive scan of hist[0..n) -> row_ptr[0..n], cursor[0..n)  (single block)
__global__ __launch_bounds__(256)
void scan_kernel(const int* __restrict__ hist,
                 int* __restrict__ row_ptr,
                 int* __restrict__ cursor, int n) {
  __shared__ int partials[256];
  const int t = threadIdx.x;
  const int chunk = (n + 255) / 256;
  int begin = t * chunk;
  int end   = begin + chunk; if (end > n) end = n;
  if (begin > n) begin = n;

  int sum = 0;
  for (int i = begin; i < end; ++i) sum += hist[i];
  partials[t] = sum;
  __syncthreads();
  if (t == 0) {
    int run = 0;
    for (int i = 0; i < 256; ++i) { int v = partials[i]; partials[i] = run; run += v; }
  }
  __syncthreads();
  int run = partials[t];
  for (int i = begin; i < end; ++i) {
    row_ptr[i] = run;
    cursor[i]  = run;
    run += hist[i];
  }
  if (end == n) row_ptr[n] = run;   // all tail threads write the same total
}

// bin edges into CSR: srcs[slot] = src[e], slots allocated per-dst via cursor
__global__ void bin_kernel(const int* __restrict__ src,
                           const int* __restrict__ dst,
                           int* __restrict__ cursor,
                           int* __restrict__ srcs, int nE) {
  int i = blockIdx.x * blockDim.x + threadIdx.x;
  if (i < nE) {
    int d = dst[i];
    int pos = atomicAdd(&cursor[d], 1);
    srcs[pos] = src[i];
  }
}

// one wave per dst node: mean of neighbor rows (256 f32), no atomics.
// lane covers 8 consecutive floats (two float4 = two b128 loads per edge).
__global__ __launch_bounds__(256)
void gather_mean_kernel(const float* __restrict__ hsrc,   // n_src x 256
                        const int* __restrict__ srcs,     // CSR neighbor list
                        const int* __restrict__ row_ptr,  // n_dst+1
                        float* __restrict__ hN,           // n_dst x 256
                        int n_dst) {
  const int d = blockIdx.x * (blockDim.x >> 5) + (threadIdx.x >> 5);
  if (d >= n_dst) return;                      // wave-uniform
  const int lane = threadIdx.x & 31;
  const int beg = row_ptr[d];
  const int end = row_ptr[d + 1];

  float4 a0 = make_float4(0.f, 0.f, 0.f, 0.f);
  float4 a1 = make_float4(0.f, 0.f, 0.f, 0.f);
  for (int j = beg; j < end; ++j) {
    int s = srcs[j];                            // wave-uniform
    const float4* p = (const float4*)(hsrc + (size_t)s * IN_F) + lane * 2;
    float4 v0 = p[0];
    float4 v1 = p[1];
    a0.x += v0.x; a0.y += v0.y; a0.z += v0.z; a0.w += v0.w;
    a1.x += v1.x; a1.y += v1.y; a1.z += v1.z; a1.w += v1.w;
  }
  const float inv = 1.0f / fmaxf((float)(end - beg), 1.0f);
  a0.x *= inv; a0.y *= inv; a0.z *= inv; a0.w *= inv;
  a1.x *= inv; a1.y *= inv; a1.z *= inv; a1.w *= inv;
  float4* o = (float4*)(hN + (size_t)d * IN_F) + lane * 2;
  o[0] = a0;
  o[1] = a1;
}

// ---------------------------------------------------------------------------
// Pack W (KDIM x N row-major) into B-fragment order.
// Per (kstep = k/4, gn = 64-col group): one 1KB block of 256 floats.
// Within a block, lane l holds 4 tile-fragments contiguously (32B):
//   float2 #(lane*4 + t) = { W[k+kh][gn*64 + t*16 + (l&15)],
//                            W[k+kh+1][same col] },  kh = (l<16) ? 0 : 2
// One thread per float2.
// ---------------------------------------------------------------------------
__global__ void pack_W_kernel(const float* __restrict__ W,
                              float* __restrict__ Bpack, int N) {
  const int ngrpN = N >> 6;
  const int total = KSTEPS * ngrpN * 128;          // float2 count
  int idx = blockIdx.x * blockDim.x + threadIdx.x;
  if (idx >= total) return;

  const int w    = idx & 127;       // slot within block
  const int t    = w & 3;           // N-tile within 64-col group
  const int lane = w >> 2;
  const int blk  = idx >> 7;
  const int gn   = blk % ngrpN;
  const int ks   = blk / ngrpN;

  const int kh  = (lane < 16) ? 0 : 2;
  const int col = gn * 64 + t * 16 + (lane & 15);
  const int kr  = ks * 4 + kh;

  float2 v;
  v.x = W[(size_t)kr * N + col];
  v.y = W[(size_t)(kr + 1) * N + col];
  ((float2*)Bpack)[idx] = v;
}

// ---------------------------------------------------------------------------
// Fused concat-GEMM:  out[M x N] = act( [A0 | A1][M x 512] @ Wpack + b )
// Each wave computes a 16x64 strip: 4 accumulators, A fragment reused 4x.
// K loop split at the concat boundary -> all loads are base + immediate.
//
// Fragment layouts (CDNA5 ISA 7.12.2, 32-bit, wave32):
//   A 16x4 : lane L<16 -> row M=L,   v0=K0, v1=K1 ; L>=16 -> row M=L-16, v0=K2, v1=K3
//   B 4x16 : lane L<16 -> col N=L,   v0=K0, v1=K1 ; L>=16 -> col N=L-16, v0=K2, v1=K3
//   C 16x16: lane L -> col N=L%16, VGPR i -> row M = i + (L<16 ? 0 : 8)
// ---------------------------------------------------------------------------
__global__ __launch_bounds__(128)
void sage_gemm_wmma_kernel(const float* __restrict__ A0,    // M x IN_F (K half 0)
                           const float* __restrict__ A1,    // M x IN_F (K half 1)
                           const float* __restrict__ Bpack, // prepacked W
                           const float* __restrict__ bias,  // N
                           float* __restrict__ out,         // M x N
                           int M, int N, int do_relu) {
  const int lane = threadIdx.x & 31;
  const int wave = threadIdx.x >> 5;
  const int ngrpN = N >> 6;                       // 64-column groups
  const int grp   = blockIdx.x * (blockDim.x >> 5) + wave;  // wave-uniform
  const int ngrps = (M >> 4) * ngrpN;
  if (grp >= ngrps) return;                       // uniform per wave

  const int tm = grp / ngrpN;
  const int gn = grp % ngrpN;

  const int l15  = lane & 15;
  const int row  = tm * 16 + l15;                 // A-fragment row
  const int col0 = gn * 64 + l15;                 // first B/C column
  const int kh   = (lane < 16) ? 0 : 2;           // half-wave K offset

  // fixed per-lane base pointers; all loop accesses are immediate offsets
  const float* Arow0 = A0 + (size_t)row * IN_F + kh;
  const float* Arow1 = A1 + (size_t)row * IN_F + kh;
  const float* Bp    = Bpack + (size_t)gn * 256 + lane * 8;
  const size_t bstride = (size_t)ngrpN * 256;     // floats per K-step

  v8f acc[4] = {};

#define K_STEP(AROW, I, KS)                                                   \
  {                                                                           \
    v2f a = *(const v2f*)((AROW) + (I) * 4);      /* b64, reused 4x */        \
    const float4* bp = (const float4*)(Bp + (size_t)(KS) * bstride);          \
    float4 bA = bp[0];                            /* tiles 0,1 (b128) */      \
    float4 bB = bp[1];                            /* tiles 2,3 (b128) */      \
    v2f b0; b0.x = bA.x; b0.y = bA.y;                                         \
    v2f b1; b1.x = bA.z; b1.y = bA.w;                                         \
    v2f b2; b2.x = bB.x; b2.y = bB.y;                                         \
    v2f b3; b3.x = bB.z; b3.y = bB.w;                                         \
    acc[0] = __builtin_amdgcn_wmma_f32_16x16x4_f32(false, a, false, b0,       \
                                                   (short)0, acc[0], false, false); \
    acc[1] = __builtin_amdgcn_wmma_f32_16x16x4_f32(false, a, false, b1,       \
                                                   (short)0, acc[1], false, false); \
    acc[2] = __builtin_amdgcn_wmma_f32_16x16x4_f32(false, a, false, b2,       \
                                                   (short)0, acc[2], false, false); \
    acc[3] = __builtin_amdgcn_wmma_f32_16x16x4_f32(false, a, false, b3,       \
                                                   (short)0, acc[3], false, false); \
  }

#pragma unroll 4
  for (int i = 0; i < KSPLIT; ++i) {              // K in [0, 256): A = A0
    K_STEP(Arow0, i, i);
  }
#pragma unroll 4
  for (int i = 0; i < KSTEPS - KSPLIT; ++i) {     // K in [256, 512): A = A1
    K_STEP(Arow1, i, KSPLIT + i);
  }
#undef K_STEP

  const int mbase = tm * 16 + ((lane < 16) ? 0 : 8);
#pragma unroll
  for (int t = 0; t < 4; ++t) {
    const int col = col0 + t * 16;
    const float bv = bias[col];
#pragma unroll
    for (int i = 0; i < 8; ++i) {
      float v = acc[t][i] + bv;
      if (do_relu) v = fmaxf(v, 0.0f);
      out[(size_t)(mbase + i) * N + col] = v;
    }
  }
}

// ---------------------------------------------------------------------------
extern "C" void kernel_launch(void* const* d_in, const int* in_sizes, int n_in,
                              void* d_out, int out_size, void* d_ws, size_t ws_size,
                              hipStream_t stream) {
  const float* x   = (const float*)d_in[0];   // 200000 x 256
  const float* W1  = (const float*)d_in[1];   // 512 x 256
  const float* b1  = (const float*)d_in[2];   // 256
  const float* W2  = (const float*)d_in[3];   // 512 x 64
  const float* b2  = (const float*)d_in[4];   // 64
  const int* src1  = (const int*)d_in[5];
  const int* dst1  = (const int*)d_in[6];
  const int* src2  = (const int*)d_in[7];
  const int* dst2  = (const int*)d_in[8];
  const int E1 = in_sizes[5];                 // 1,000,000
  const int E2 = in_sizes[7];                 // 200,000

  // workspace layout (4-byte units)
  char* wsb = (char*)d_ws;
  size_t off = 0;
  auto take = [&](size_t n_elems) { void* p = wsb + off; off += n_elems * 4; return p; };

  int* hist1    = (int*)take(N_DST1);          // zeroed each call
  int* hist2    = (int*)take(N_DST2);          // (contiguous with hist1)
  int* row_ptr1 = (int*)take(N_DST1 + 1);
  int* cursor1  = (int*)take(N_DST1);
  int* srcs1    = (int*)take((size_t)E1);
  int* row_ptr2 = (int*)take(N_DST2 + 1);
  int* cursor2  = (int*)take(N_DST2);
  int* srcs2    = (int*)take((size_t)E2);
  float* hN1    = (float*)take((size_t)N_DST1 * IN_F);
  float* h      = (float*)take((size_t)N_DST1 * HID_F);
  float* hN2    = (float*)take((size_t)N_DST2 * IN_F);
  float* Bpack1 = (float*)take((size_t)KDIM * HID_F);   // 512 KB, repacked W1
  float* Bpack2 = (float*)take((size_t)KDIM * N_CLS);   // 128 KB, repacked W2
  (void)ws_size;

  // ---- weight prepacking (cheap, once per launch) ----
  {
    int t1 = KSTEPS * (HID_F / 64) * 128;   // 65,536 float2 slots
    int t2 = KSTEPS * (N_CLS / 64) * 128;   // 16,384
    pack_W_kernel<<<(t1 + 255) / 256, 256, 0, stream>>>(W1, Bpack1, HID_F);
    pack_W_kernel<<<(t2 + 255) / 256, 256, 0, stream>>>(W2, Bpack2, N_CLS);
  }

  // ---- layer 1: CSR build + gather-mean ----
  zero_i32_kernel<<<(N_DST1 + N_DST2 + 255) / 256, 256, 0, stream>>>(
      hist1, N_DST1 + N_DST2);
  hist_kernel<<<(E1 + 255) / 256, 256, 0, stream>>>(dst1, hist1, E1);
  scan_kernel<<<1, 256, 0, stream>>>(hist1, row_ptr1, cursor1, N_DST1);
  bin_kernel<<<(E1 + 255) / 256, 256, 0, stream>>>(src1, dst1, cursor1, srcs1, E1);
  gather_mean_kernel<<<N_DST1 / 8, 256, 0, stream>>>(x, srcs1, row_ptr1, hN1, N_DST1);

  // h = relu([x[:50000] | hN1] @ W1 + b1)   (M=50000, N=256)
  {
    int ngrps = (N_DST1 / 16) * (HID_F / 64);   // 12,500 wave-strips
    sage_gemm_wmma_kernel<<<ngrps / 4, 128, 0, stream>>>(
        x, hN1, Bpack1, b1, h, N_DST1, HID_F, /*relu=*/1);
  }

  // ---- layer 2: CSR build + gather-mean over h ----
  hist_kernel<<<(E2 + 255) / 256, 256, 0, stream>>>(dst2, hist2, E2);
  scan_kernel<<<1, 256, 0, stream>>>(hist2, row_ptr2, cursor2, N_DST2);
  bin_kernel<<<(E2 + 255) / 256, 256, 0, stream>>>(src2, dst2, cursor2, srcs2, E2);
  gather_mean_kernel<<<N_DST2 / 8, 256, 0, stream>>>(h, srcs2, row_ptr2, hN2, N_DST2);

  // out = [h[:10000] | hN2] @ W2 + b2        (M=10000, N=64)
  {
    int ngrps = (N_DST2 / 16) * (N_CLS / 64);   // 625 wave-strips
    sage_gemm_wmma_kernel<<<(ngrps + 3) / 4, 128, 0, stream>>>(
        h, hN2, Bpack2, b2, (float*)d_out, N_DST2, N_CLS, /*relu=*/0);
  }
}